// BoundaryLoss_60224031424719
// MI455X (gfx1250) — compile-verified
//
#include <hip/hip_runtime.h>

#define Bn 2
#define Cn 4
#define Dn 96
#define Hn 96
#define Wn 96
#define VOL  (Dn*Hn*Wn)        // 884736 voxels per batch item
#define NVOX (Bn*VOL)          // 1769472 total voxels
#define NLINES (Bn*96*96)      // 18432 1-D lines per EDT pass
#define BIGF 1e10f
#define INV_THETA 0.2f         // 1/5.0

typedef __attribute__((ext_vector_type(2))) float v2f;
typedef __attribute__((ext_vector_type(8))) float v8f;

// -------- 1) boundary mask -> squared-distance seed (0 on boundary, BIG elsewhere)
__global__ void bl_boundary_init(const int* __restrict__ tgt, float* __restrict__ f) {
    int i = blockIdx.x * blockDim.x + threadIdx.x;
    if (i >= NVOX) return;
    int r  = i % VOL;
    int z  = r / (Hn * Wn);
    int r2 = r % (Hn * Wn);
    int y  = r2 / Wn;
    int x  = r2 % Wn;
    int t = tgt[i];
    bool bd = false;
    if (z + 1 < Dn) bd |= (tgt[i + Hn * Wn] != t);
    if (y + 1 < Hn) bd |= (tgt[i + Wn] != t);
    if (x + 1 < Wn) bd |= (tgt[i + 1] != t);
    f[i] = bd ? 0.0f : BIGF;
}

// -------- 2a) tiled EDT pass for STRIDED axes (D: se=9216, H: se=96).
// Tile = 96 line-elements x 32 contiguous-x lines staged in LDS (12 KB).
// Global loads/stores: 32 lanes touch 128 contiguous bytes -> fully coalesced.
// LDS compute reads tile[j*32 + lane] -> lane-consecutive, conflict-free (64 banks).
__global__ void bl_edt_pass_tiled(const float* __restrict__ fin, float* __restrict__ fout,
                                  int se, int sw) {
    __shared__ float tile[96 * 32];
    int bi  = blockIdx.x;               // 0 .. Bn*96*3-1
    int b   = bi / (96 * 3);
    int rem = bi % (96 * 3);
    int w   = rem / 3;                  // the other (non-line, non-x) coordinate
    int x0  = (rem % 3) * 32;           // x-chunk
    long base = (long)b * VOL + (long)w * sw + x0;

    #pragma unroll
    for (int k = 0; k < 12; ++k) {
        int flat = threadIdx.x + 256 * k;   // 0..3071
        int e    = flat >> 5;               // element along line, 0..95
        int xi   = flat & 31;
        tile[flat] = fin[base + (long)e * se + xi];
    }
    __syncthreads();

    for (int k = 0; k < 12; ++k) {
        int flat = threadIdx.x + 256 * k;
        int i    = flat >> 5;
        int xi   = flat & 31;
        float m = 3.0e38f;
        #pragma unroll 8
        for (int j = 0; j < 96; ++j) {
            float d = (float)(i - j);
            m = fminf(m, tile[(j << 5) + xi] + d * d);  // single fp32 add + exact min
        }
        fout[base + (long)i * se + xi] = m;
    }
}

// -------- 2b) per-line EDT pass for the CONTIGUOUS axis (W, stride 1).
__global__ void bl_edt_pass(const float* __restrict__ fin, float* __restrict__ fout,
                            int stride, int su, int sv) {
    __shared__ float line[96];
    int l = blockIdx.x;              // 0 .. NLINES-1
    int b = l / (96 * 96);
    int r = l % (96 * 96);
    int u = r / 96;
    int v = r % 96;
    long base = (long)b * VOL + (long)u * su + (long)v * sv;
    int i = threadIdx.x;             // 0..95
    line[i] = fin[base + (long)i * stride];
    __syncthreads();
    float m = 3.0e38f;
    #pragma unroll
    for (int j = 0; j < 96; ++j) {
        float d = (float)(i - j);
        m = fminf(m, line[j] + d * d);
    }
    fout[base + (long)i * stride] = m;
}

// -------- 3) zero the scalar output
__global__ void bl_zero(float* __restrict__ out) {
    if (threadIdx.x == 0 && blockIdx.x == 0) out[0] = 0.0f;
}

// -------- 4) softmax + weighted SSE + WMMA-f32 wave reduction + one atomic/wave
__global__ void bl_loss(const float* __restrict__ pred, const int* __restrict__ tgt,
                        const float* __restrict__ dist2, float* __restrict__ out) {
    int tid = blockIdx.x * blockDim.x + threadIdx.x;
    int nthreads = gridDim.x * blockDim.x;
    float partial = 0.0f;
    for (int i = tid; i < NVOX; i += nthreads) {
        int b = i / VOL;
        int r = i % VOL;
        const float* p = pred + (long)b * (Cn * VOL) + r;
        float p0 = p[0 * VOL], p1 = p[1 * VOL], p2 = p[2 * VOL], p3 = p[3 * VOL];
        float mx = fmaxf(fmaxf(p0, p1), fmaxf(p2, p3));
        float e0 = expf(p0 - mx), e1 = expf(p1 - mx), e2 = expf(p2 - mx), e3 = expf(p3 - mx);
        float inv = 1.0f / (e0 + e1 + e2 + e3);
        float q0 = e0 * inv, q1 = e1 * inv, q2 = e2 * inv, q3 = e3 * inv;
        int t = tgt[i];
        q0 -= (t == 0) ? 1.0f : 0.0f;
        q1 -= (t == 1) ? 1.0f : 0.0f;
        q2 -= (t == 2) ? 1.0f : 0.0f;
        q3 -= (t == 3) ? 1.0f : 0.0f;
        float sse = q0 * q0 + q1 * q1 + q2 * q2 + q3 * q3;
        float w = expf(-sqrtf(dist2[i]) * INV_THETA);
        partial += w * sse;
    }
    partial *= (1.0f / (float)NVOX);

    // Wave reduction on the matrix unit, fully fp32 (no precision loss):
    // A(16x4): lane L<16 -> A[L][0]=partial; lane L>=16 -> A[L-16][2]=partial; B = ones.
    // D[m][n] = partial_m + partial_{m+16}; lane n<16 holds rows 0..7, lane n>=16 rows 8..15.
    // component-sum(s): lanes 0..15 = S_low, lanes 16..31 = S_high; total = s + shfl_xor(s,16).
    v2f a;  a[0] = partial; a[1] = 0.0f;
    v2f bb; bb[0] = 1.0f;   bb[1] = 1.0f;
    v8f c = {};
    v8f d = __builtin_amdgcn_wmma_f32_16x16x4_f32(
        /*neg_a=*/false, a, /*neg_b=*/false, bb,
        /*c_mod=*/(short)0, c, /*reuse_a=*/false, /*reuse_b=*/false);
    float s = d[0] + d[1] + d[2] + d[3] + d[4] + d[5] + d[6] + d[7];
    float total = s + __shfl_xor(s, 16, 32);
    if ((threadIdx.x & 31) == 0) atomicAdd(out, total);
}

extern "C" void kernel_launch(void* const* d_in, const int* in_sizes, int n_in,
                              void* d_out, int out_size, void* d_ws, size_t ws_size,
                              hipStream_t stream) {
    const float* pred = (const float*)d_in[0];
    const int*   tgt  = (const int*)d_in[1];
    float* out  = (float*)d_out;
    float* bufA = (float*)d_ws;          // NVOX floats
    float* bufB = bufA + NVOX;           // NVOX floats

    bl_boundary_init<<<(NVOX + 255) / 256, 256, 0, stream>>>(tgt, bufA);

    // EDT pass along D (element stride 9216): tiled, w = y (sw = 96)
    bl_edt_pass_tiled<<<Bn * 96 * 3, 256, 0, stream>>>(bufA, bufB, Hn * Wn, Wn);
    // EDT pass along H (element stride 96): tiled, w = z (sw = 9216)
    bl_edt_pass_tiled<<<Bn * 96 * 3, 256, 0, stream>>>(bufB, bufA, Wn, Hn * Wn);
    // EDT pass along W (contiguous lines): per-line kernel
    bl_edt_pass<<<NLINES, 96, 0, stream>>>(bufA, bufB, 1, Hn * Wn, Wn);

    bl_zero<<<1, 32, 0, stream>>>(out);
    bl_loss<<<1024, 256, 0, stream>>>(pred, tgt, bufB, out);
}